// GCN_14027363188818
// MI455X (gfx1250) — compile-verified
//
#include <hip/hip_runtime.h>

typedef __attribute__((ext_vector_type(2))) float v2f;
typedef __attribute__((ext_vector_type(8))) float v8f;

// ---------------------------------------------------------------------------
// Degree / normalization
// ---------------------------------------------------------------------------
__global__ void gcn_deg_count(const int* __restrict__ dst, float* __restrict__ deg, int E) {
    int i = blockIdx.x * blockDim.x + threadIdx.x;
    if (i < E) atomicAdd(&deg[dst[i]], 1.0f);
}

__global__ void gcn_deg_to_dinv(float* __restrict__ deg, int N) {
    int i = blockIdx.x * blockDim.x + threadIdx.x;
    if (i < N) deg[i] = rsqrtf(deg[i] + 1.0f);   // deg >= 1 -> finite
}

// ---------------------------------------------------------------------------
// Fused dense layer:
//   H[N,NOUT]   = act(X)[N,K] * W[K,NOUT]        (act = leaky-relu 0.2 or id)
//   AGG[N,NOUT] = H * dinv^2 + bias              (self-loop + bias; buffer init)
// via V_WMMA_F32_16X16X4_F32, one wave per 16x16 tile, 256-thread blocks
// (8 waves sharing one 16-col W tile staged in LDS).
//
// Fragment layouts per CDNA5 ISA 7.12.2:
//  A (16x4 f32):  lanes 0-15 row m: VGPR0=K+0 VGPR1=K+1; lanes 16-31: K+2/K+3
//  B (4x16 f32):  VGPR0: lanes0-15 K+0, lanes16-31 K+2; VGPR1: K+1 / K+3
//  C/D (16x16):   VGPR r: lanes0-15 M=r, lanes16-31 M=r+8; N = lane&15
//
// LDS: W tile transposed [n][k], row padded by 4 dwords -> b.x/b.y are one
// 8-byte ds_load_b64; banks: lanes0-15 hit (4m+k)%64 / +1, lanes16-31 hit
// (4m+k+2)%64 / +3 -> disjoint mod 4, conflict-free.
// ---------------------------------------------------------------------------
template <int K, int NOUT, bool LEAKY>
__global__ void gcn_gemm_wmma(const float* __restrict__ X, const float* __restrict__ W,
                              const float* __restrict__ bias, const float* __restrict__ dinv,
                              float* __restrict__ H, float* __restrict__ AGG, int nRowTiles) {
    constexpr int KP = K + 4;                    // padded LDS row (conflict-free)
    __shared__ float ldsT[16 * KP];

    const int tid  = threadIdx.x;
    const int col0 = blockIdx.y * 16;

    // Cooperative B-tile fill: coalesced global reads, conflict-free LDS writes
    {
        const int n = tid & 15;
        for (int k = tid >> 4; k < K; k += 16)   // blockDim.x == 256
            ldsT[n * KP + k] = W[(size_t)k * NOUT + col0 + n];
    }
    __syncthreads();

    const int lane    = tid & 31;
    const int wave    = tid >> 5;
    const int rowTile = blockIdx.x * 8 + wave;
    if (rowTile >= nRowTiles) return;            // wave-uniform branch

    const int m  = lane & 15;                    // row-in-tile (A) / col-in-tile (B, C)
    const int hi = lane >> 4;                    // selects K pair {0,1} vs {2,3}
    const int row0 = rowTile * 16;

    const float* xrow = X + (size_t)(row0 + m) * K;
    __builtin_prefetch(xrow, 0, 3);              // -> global_prefetch_b8

    v8f acc = {};
#pragma unroll 8
    for (int k = 0; k < K; k += 4) {
        v2f a, b;
        float2 av = *(const float2*)(xrow + k + hi * 2);   // 8B aligned pair
        if constexpr (LEAKY) {                   // activation fused into A-load
            av.x = av.x > 0.0f ? av.x : 0.2f * av.x;
            av.y = av.y > 0.0f ? av.y : 0.2f * av.y;
        }
        a.x = av.x; a.y = av.y;
        b = *(const v2f*)&ldsT[m * KP + k + hi * 2];       // ds_load_b64
        acc = __builtin_amdgcn_wmma_f32_16x16x4_f32(
            /*neg_a=*/false, a, /*neg_b=*/false, b,
            /*c_mod=*/(short)0, acc, /*reuse_a=*/false, /*reuse_b=*/false);
    }

    // Fused epilogue: write H and AGG = H*dinv^2 + bias
    const float bcol = bias[col0 + m];
    float* hrow = H   + (size_t)(row0 + hi * 8) * NOUT + (col0 + m);
    float* arow = AGG + (size_t)(row0 + hi * 8) * NOUT + (col0 + m);
#pragma unroll
    for (int r = 0; r < 8; ++r) {
        const float di = dinv[row0 + hi * 8 + r];
        const float hv = acc[r];
        hrow[(size_t)r * NOUT] = hv;
        arow[(size_t)r * NOUT] = hv * di * di + bcol;
    }
}

// ---------------------------------------------------------------------------
// Edge scatter: one wave per edge; lane owns F/32 contiguous features.
// agg[dst] += h[src] * dinv[src]*dinv[dst]   via global_atomic_add_f32
// ---------------------------------------------------------------------------
template <int F>
__global__ void gcn_scatter(const int* __restrict__ src, const int* __restrict__ dst,
                            const float* __restrict__ dinv, const float* __restrict__ H,
                            float* __restrict__ AGG, int E) {
    const int lane = threadIdx.x & 31;
    const int e    = (int)(((long long)blockIdx.x * blockDim.x + threadIdx.x) >> 5);
    if (e >= E) return;
    const int s = src[e], d = dst[e];
    const float w = dinv[s] * dinv[d];
    constexpr int PER = F / 32;                  // 4 (F=128) or 2 (F=64)
    const float* hs = H + (size_t)s * F + lane * PER;
    float* ad = AGG + (size_t)d * F + lane * PER;
    if constexpr (PER == 4) {
        float4 v = *(const float4*)hs;
        atomicAdd(&ad[0], v.x * w);
        atomicAdd(&ad[1], v.y * w);
        atomicAdd(&ad[2], v.z * w);
        atomicAdd(&ad[3], v.w * w);
    } else {
        float2 v = *(const float2*)hs;
        atomicAdd(&ad[0], v.x * w);
        atomicAdd(&ad[1], v.y * w);
    }
}

// ---------------------------------------------------------------------------
// Host-side orchestration
// ---------------------------------------------------------------------------
static inline unsigned cdiv(long long a, long long b) { return (unsigned)((a + b - 1) / b); }

extern "C" void kernel_launch(void* const* d_in, const int* in_sizes, int n_in,
                              void* d_out, int out_size, void* d_ws, size_t ws_size,
                              hipStream_t stream) {
    (void)n_in; (void)out_size; (void)ws_size;
    const float* x  = (const float*)d_in[0];
    const int*   ei = (const int*)d_in[1];        // int32 (JAX x64 disabled)
    const float* W1 = (const float*)d_in[2];
    const float* b1 = (const float*)d_in[3];
    const float* W2 = (const float*)d_in[4];
    const float* b2 = (const float*)d_in[5];
    const float* W3 = (const float*)d_in[6];
    const float* b3 = (const float*)d_in[7];

    const int N = in_sizes[0] / 512;              // 50000
    const int E = in_sizes[1] / 2;                // 800000
    const int* src = ei;
    const int* dst = ei + E;

    // Workspace: dinv | B0 = H (N*128) | B1 = AGG1/X2 (N*128) | B2 = AGG2/X3 (N*64)
    char* ws = (char*)d_ws;
    size_t offB0 = (((size_t)N * 4) + 255) & ~(size_t)255;
    size_t offB1 = offB0 + (size_t)N * 128 * 4;
    size_t offB2 = offB1 + (size_t)N * 128 * 4;
    float* dinv = (float*)ws;
    float* B0   = (float*)(ws + offB0);
    float* B1   = (float*)(ws + offB1);
    float* B2   = (float*)(ws + offB2);
    float* out  = (float*)d_out;

    const int nRowTiles = (N + 15) / 16;          // 3125 (exact for N=50000)
    const unsigned gX = cdiv(nRowTiles, 8);       // 8 waves / 256-thread block

    // --- degree normalization ---
    hipMemsetAsync(dinv, 0, (size_t)N * 4, stream);
    gcn_deg_count<<<cdiv(E, 256), 256, 0, stream>>>(dst, dinv, E);
    gcn_deg_to_dinv<<<cdiv(N, 256), 256, 0, stream>>>(dinv, N);

    // --- layer 1: 512 -> 128 : H1=B0, AGG1=B1 ---
    gcn_gemm_wmma<512, 128, false><<<dim3(gX, 8), 256, 0, stream>>>(x, W1, b1, dinv, B0, B1, nRowTiles);
    gcn_scatter<128><<<cdiv((long long)E * 32, 256), 256, 0, stream>>>(src, dst, dinv, B0, B1, E);

    // --- layer 2: 128 -> 64 : X2=leaky(B1) fused, H2=B0, AGG2=B2 ---
    gcn_gemm_wmma<128, 64, true><<<dim3(gX, 4), 256, 0, stream>>>(B1, W2, b2, dinv, B0, B2, nRowTiles);
    gcn_scatter<64><<<cdiv((long long)E * 32, 256), 256, 0, stream>>>(src, dst, dinv, B0, B2, E);

    // --- layer 3: 64 -> 64 : X3=leaky(B2) fused, H3=B0, AGG3=d_out ---
    gcn_gemm_wmma<64, 64, true><<<dim3(gX, 4), 256, 0, stream>>>(B2, W3, b3, dinv, B0, out, nRowTiles);
    gcn_scatter<64><<<cdiv((long long)E * 32, 256), 256, 0, stream>>>(src, dst, dinv, B0, out, E);
}